// NormDecoderBlock_49709951484782
// MI455X (gfx1250) — compile-verified
//
#include <hip/hip_runtime.h>
#include <math.h>

#define DIMC   256
#define PROJC  128
#define NPOSC  8
#define EXTC   64
#define NLAY   4
#define NHEADS 4
#define KDIMC  136   // PROJ + NPOS
#define HDC    64    // DIM / HEADS

typedef __attribute__((ext_vector_type(2))) float v2f;
typedef __attribute__((ext_vector_type(8))) float v8f;

__device__ __forceinline__ float wave_sum(float v) {
#pragma unroll
  for (int off = 16; off > 0; off >>= 1) v += __shfl_xor(v, off, 32);
  return v;
}
__device__ __forceinline__ float wave_max(float v) {
#pragma unroll
  for (int off = 16; off > 0; off >>= 1) v = fmaxf(v, __shfl_xor(v, off, 32));
  return v;
}

// ---------------------------------------------------------------------------
// src = concat([input, pos], axis=1)  : (L, KDIM)
// pos[n, i] = mod(n, 2^(i+1)) / 2^i
// ---------------------------------------------------------------------------
__global__ void k_build_src(const float* __restrict__ inp, float* __restrict__ src, int L) {
  int idx = blockIdx.x * blockDim.x + threadIdx.x;
  if (idx >= L * KDIMC) return;
  int row = idx / KDIMC;
  int c   = idx % KDIMC;
  float v;
  if (c < PROJC) {
    v = inp[row * PROJC + c];
  } else {
    int i = c - PROJC;
    float two_i = (float)(1u << i);
    v = fmodf((float)row, 2.0f * two_i) / two_i;
  }
  src[idx] = v;
}

// ---------------------------------------------------------------------------
// fp32 WMMA GEMM: C[m][n] = sum_k A[m][k] * W[n][k] + bias[n] (+ add[m][n])
// A: (M,K) row-major, W: (N,K) row-major. M % 128 == 0, N % 16 == 0, K % 8 == 0.
// block = (32, 8): 8 waves tile along M; the shared 16xK W tile is staged in
// LDS once per block (coalesced) and served to all waves via ds_load.
// Two independent accumulators keep >=2 WMMAs in flight (breaks the C-operand
// RAW chain of V_WMMA_F32_16X16X4_F32).
// LDS layout ldsW[k][n] (row stride 16 floats): lanes 0-15 hit banks
// 16k..16k+15, lanes 16-31 (row k+2) hit the group offset by 32 -> no conflict.
// ---------------------------------------------------------------------------
__global__ void k_gemm_wmma(const float* __restrict__ A, const float* __restrict__ W,
                            const float* __restrict__ bias, const float* __restrict__ add,
                            float* __restrict__ C, int M, int N, int K) {
  __shared__ float ldsW[256 * 16];            // [k][n], K <= 256
  const int tid = threadIdx.y * 32 + threadIdx.x;
  const int n0  = blockIdx.x * 16;

  for (int idx = tid; idx < 16 * K; idx += 256) {
    int n = idx / K;
    int k = idx - n * K;
    ldsW[k * 16 + n] = W[(size_t)(n0 + n) * K + k];
  }
  __syncthreads();

  const int lane = threadIdx.x;
  const int m0   = (blockIdx.y * blockDim.y + threadIdx.y) * 16;
  const int l15  = lane & 15;
  const int hi   = lane >> 4;

  float bval = bias[n0 + l15];
  v8f acc0, acc1;
#pragma unroll
  for (int r = 0; r < 8; ++r) { acc0[r] = bval; acc1[r] = 0.0f; }

  const float* arow = A + (size_t)(m0 + l15) * K + hi * 2;
  const float* wk   = ldsW + (hi * 2) * 16 + l15;   // + k*16 per K step

  for (int k0 = 0; k0 < K; k0 += 8) {
    __builtin_prefetch(&arow[k0 + 8], 0, 0);        // global_prefetch_b8, next chunk
    v2f a0, b0, a1, b1;
    a0.x = arow[k0];
    a0.y = arow[k0 + 1];
    b0.x = wk[(k0)     * 16];
    b0.y = wk[(k0 + 1) * 16];
    acc0 = __builtin_amdgcn_wmma_f32_16x16x4_f32(false, a0, false, b0,
                                                 (short)0, acc0, false, false);
    a1.x = arow[k0 + 4];
    a1.y = arow[k0 + 5];
    b1.x = wk[(k0 + 4) * 16];
    b1.y = wk[(k0 + 5) * 16];
    acc1 = __builtin_amdgcn_wmma_f32_16x16x4_f32(false, a1, false, b1,
                                                 (short)0, acc1, false, false);
  }

#pragma unroll
  for (int r = 0; r < 8; ++r) {
    int m = m0 + hi * 8 + r;
    size_t idx = (size_t)m * N + n0 + l15;
    float v = acc0[r] + acc1[r];
    if (add) v += add[idx];
    C[idx] = v;
  }
}

// ---------------------------------------------------------------------------
// 1D conv (k=3, pad=1) over time, DIM->DIM channels, + bias, tanh.
// Y[t][co] = tanh( b[co] + sum_{dt,ci} Wc[co][ci][dt] * X[t+dt-1][ci] )
// 3 accumulated WMMA GEMMs; all three 16x256 weight taps staged in LDS (48KB).
// ---------------------------------------------------------------------------
__global__ void k_conv_wmma(const float* __restrict__ X, const float* __restrict__ Wc,
                            const float* __restrict__ bc, float* __restrict__ Y, int T) {
  __shared__ float ldsW[3 * 256 * 16];        // [dt][ci][n]
  const int tid = threadIdx.y * 32 + threadIdx.x;
  const int n0  = blockIdx.x * 16;

  for (int idx = tid; idx < 3 * 256 * 16; idx += 256) {
    int n   = idx / (3 * 256);
    int rem = idx - n * (3 * 256);
    int ci  = rem / 3;
    int dt  = rem - ci * 3;                   // coalesced read of Wc[n0+n][*][*]
    ldsW[(dt * 256 + ci) * 16 + n] = Wc[(size_t)(n0 + n) * (DIMC * 3) + rem];
  }
  __syncthreads();

  const int lane = threadIdx.x;
  const int m0   = (blockIdx.y * blockDim.y + threadIdx.y) * 16;
  const int l15  = lane & 15;
  const int hi   = lane >> 4;

  float bval = bc[n0 + l15];
  v8f acc0, acc1;
#pragma unroll
  for (int r = 0; r < 8; ++r) { acc0[r] = bval; acc1[r] = 0.0f; }

#pragma unroll
  for (int dt = 0; dt < 3; ++dt) {
    int  row = m0 + l15 + dt - 1;
    bool inb = (row >= 0) && (row < T);
    const float* arow = X + (size_t)row * DIMC + hi * 2;     // not deref'd if !inb
    const float* wk   = ldsW + (dt * 256 + hi * 2) * 16 + l15;
    for (int k0 = 0; k0 < DIMC; k0 += 8) {
      v2f a0, b0, a1, b1;
      a0.x = inb ? arow[k0]     : 0.0f;
      a0.y = inb ? arow[k0 + 1] : 0.0f;
      b0.x = wk[(k0)     * 16];
      b0.y = wk[(k0 + 1) * 16];
      acc0 = __builtin_amdgcn_wmma_f32_16x16x4_f32(false, a0, false, b0,
                                                   (short)0, acc0, false, false);
      a1.x = inb ? arow[k0 + 4] : 0.0f;
      a1.y = inb ? arow[k0 + 5] : 0.0f;
      b1.x = wk[(k0 + 4) * 16];
      b1.y = wk[(k0 + 5) * 16];
      acc1 = __builtin_amdgcn_wmma_f32_16x16x4_f32(false, a1, false, b1,
                                                   (short)0, acc1, false, false);
    }
  }

#pragma unroll
  for (int r = 0; r < 8; ++r) {
    int m = m0 + hi * 8 + r;
    Y[(size_t)m * DIMC + n0 + l15] = tanhf(acc0[r] + acc1[r]);
  }
}

// ---------------------------------------------------------------------------
// Banded multihead attention. One wave per (query row j, head h).
// Allowed keys: -EXT <= j - 5i < 5 + EXT  ->  i in [ceil((j-68)/5), (j+64)/5],
// at most 28 keys -> one key per lane, single pass, shfl softmax.
// ---------------------------------------------------------------------------
__global__ void k_attn(const float* __restrict__ Q, const float* __restrict__ Kb,
                       const float* __restrict__ Vb, float* __restrict__ Ctx,
                       int T, int L) {
  const int j    = blockIdx.x;
  const int h    = blockIdx.y;
  const int lane = threadIdx.x;

  const float* qr = Q + (size_t)j * DIMC + h * HDC;

  int i_lo = (j >= 68) ? ((j - 68 + 4) / 5) : 0;
  int i_hi = (j + EXTC) / 5;
  if (i_hi > L - 1) i_hi = L - 1;

  int  i   = i_lo + lane;
  bool act = (i <= i_hi);

  float score = -1e30f;
  if (act) {
    const float* kr = Kb + (size_t)i * DIMC + h * HDC;
    float s = 0.0f;
#pragma unroll 8
    for (int d = 0; d < HDC; ++d) s += qr[d] * kr[d];
    score = s * 0.125f;                  // 1/sqrt(64)
  }

  float mx  = wave_max(score);
  float p   = act ? __expf(score - mx) : 0.0f;
  float den = wave_sum(p);
  float inv = 1.0f / den;

  const float* vr = Vb + (size_t)(act ? i : 0) * DIMC + h * HDC;
  for (int d = 0; d < HDC; ++d) {
    float c = act ? p * vr[d] : 0.0f;
    c = wave_sum(c);
    if (lane == 0) Ctx[(size_t)j * DIMC + h * HDC + d] = c * inv;
  }
}

// ---------------------------------------------------------------------------
// Row LayerNorm over 256 (no affine), optional residual add: Y = LN(X) (+ R)
// One wave per row, 8 elements per lane.
// ---------------------------------------------------------------------------
__global__ void k_ln_add(const float* __restrict__ X, const float* __restrict__ R,
                         float* __restrict__ Y) {
  const int row  = blockIdx.x;
  const int lane = threadIdx.x;
  const size_t base = (size_t)row * DIMC;

  float v[8];
  float s = 0.0f;
#pragma unroll
  for (int r = 0; r < 8; ++r) { v[r] = X[base + lane + r * 32]; s += v[r]; }
  float mu = wave_sum(s) * (1.0f / DIMC);

  float var = 0.0f;
#pragma unroll
  for (int r = 0; r < 8; ++r) { float d = v[r] - mu; var += d * d; }
  var = wave_sum(var) * (1.0f / DIMC);
  float rstd = rsqrtf(var + 1e-5f);

#pragma unroll
  for (int r = 0; r < 8; ++r) {
    float o = (v[r] - mu) * rstd;
    if (R) o += R[base + lane + r * 32];
    Y[base + lane + r * 32] = o;
  }
}

// ---------------------------------------------------------------------------
extern "C" void kernel_launch(void* const* d_in, const int* in_sizes, int n_in,
                              void* d_out, int out_size, void* d_ws, size_t ws_size,
                              hipStream_t stream) {
  (void)n_in; (void)out_size; (void)ws_size;
  const float* input    = (const float*)d_in[0];
  const float* residual = (const float*)d_in[1];
  const float* W_proj   = (const float*)d_in[2];
  const float* b_proj   = (const float*)d_in[3];
  const float* Wq       = (const float*)d_in[4];
  const float* bq       = (const float*)d_in[5];
  const float* Wk       = (const float*)d_in[6];
  const float* bk       = (const float*)d_in[7];
  const float* Wv       = (const float*)d_in[8];
  const float* bvp      = (const float*)d_in[9];
  const float* Wo       = (const float*)d_in[10];
  const float* bo       = (const float*)d_in[11];
  const float* conv_w   = (const float*)d_in[12];
  const float* conv_b   = (const float*)d_in[13];
  const float* W_skip   = (const float*)d_in[14];
  const float* b_skip   = (const float*)d_in[15];
  float* out = (float*)d_out;

  const int L = in_sizes[0] / PROJC;   // 2048
  const int T = 5 * L;                 // 10240

  // Workspace layout (floats): ~36.8 MB total, fits L2.
  float* ws  = (float*)d_ws;
  size_t o   = 0;
  float* src = ws + o; o += (size_t)L * KDIMC;  // (L,136)
  float* kb  = ws + o; o += (size_t)L * DIMC;   // (L,256)
  float* vb  = ws + o; o += (size_t)L * DIMC;   // (L,256)
  float* B1  = ws + o; o += (size_t)T * DIMC;   // tgt -> ctx -> cnn_in
  float* B2  = ws + o; o += (size_t)T * DIMC;   // q -> attn_out -> conv ping -> skip pre
  float* B3  = ws + o; o += (size_t)T * DIMC;   // conv pong (cnn_out)

  dim3 wb(32, 8);  // 8 wave32 waves / block

  // 1) src = [input | pos]
  k_build_src<<<(L * KDIMC + 255) / 256, 256, 0, stream>>>(input, src, L);

  // 2) tgt = src @ W_proj^T + b_proj : (L,1280) viewed flat as (T,256) -> B1
  k_gemm_wmma<<<dim3((5 * DIMC) / 16, L / 128), wb, 0, stream>>>(
      src, W_proj, b_proj, nullptr, B1, L, 5 * DIMC, KDIMC);

  // 3) q = tgt @ Wq^T + bq -> B2
  k_gemm_wmma<<<dim3(DIMC / 16, T / 128), wb, 0, stream>>>(
      B1, Wq, bq, nullptr, B2, T, DIMC, DIMC);

  // 4) k,v = src @ {Wk,Wv}^T + {bk,bv}
  k_gemm_wmma<<<dim3(DIMC / 16, L / 128), wb, 0, stream>>>(
      src, Wk, bk, nullptr, kb, L, DIMC, KDIMC);
  k_gemm_wmma<<<dim3(DIMC / 16, L / 128), wb, 0, stream>>>(
      src, Wv, bvp, nullptr, vb, L, DIMC, KDIMC);

  // 5) banded attention: ctx -> B1 (tgt dead after q)
  k_attn<<<dim3(T, NHEADS), 32, 0, stream>>>(B2, kb, vb, B1, T, L);

  // 6) attn_out = ctx @ Wo^T + bo -> B2 (q dead)
  k_gemm_wmma<<<dim3(DIMC / 16, T / 128), wb, 0, stream>>>(
      B1, Wo, bo, nullptr, B2, T, DIMC, DIMC);

  // 7) cnn_in = LN(attn_out) + residual -> B1
  k_ln_add<<<T, 32, 0, stream>>>(B2, residual, B1);

  // 8) conv stack (4 layers, tanh), ping-pong B2/B3
  const float* cin = B1;
  float* cout = B2;
  for (int l = 0; l < NLAY; ++l) {
    k_conv_wmma<<<dim3(DIMC / 16, T / 128), wb, 0, stream>>>(
        cin, conv_w + (size_t)l * DIMC * DIMC * 3, conv_b + (size_t)l * DIMC, cout, T);
    if (l == 0) { cin = B2; cout = B3; }
    else        { const float* t = cin; cin = cout; cout = (float*)t; }
  }
  // after loop: cin == B3 == cnn_out, B2 free

  // 9) pre = cnn_in @ W_skip^T + b_skip + cnn_out -> B2
  k_gemm_wmma<<<dim3(DIMC / 16, T / 128), wb, 0, stream>>>(
      B1, W_skip, b_skip, cin, B2, T, DIMC, DIMC);

  // 10) out = LN(pre)
  k_ln_add<<<T, 32, 0, stream>>>(B2, nullptr, out);
}